// SelfDWConv2d_5454608465980
// MI455X (gfx1250) — compile-verified
//
#include <hip/hip_runtime.h>
#include <hip/hip_bf16.h>

// Problem constants (from reference)
#define BB 32
#define CC 192
#define HH 56
#define WW 56
#define KK 7
#define PC (CC * KK * KK)   // 9408 elements per-sample for LayerNorm
#define LN_EPS 1e-5f

// img LDS row stride: 76 floats = 304 B. 76 mod 64 = 12, and 12*m mod 64 is
// distinct for m=0..15 -> A-fragment column reads (16 consecutive rows, same
// col) are bank-conflict-free; 304 B is 8B-aligned so paired b64 DS loads of
// (col, col+1) with even col stay natural-aligned.
#define IMG_W 76

typedef float v2f  __attribute__((ext_vector_type(2)));
typedef float v8f  __attribute__((ext_vector_type(8)));

// ---------------------------------------------------------------------------
// Kernel 1: dynamic-kernel generation.
// One block per batch sample b (LayerNorm normalizes over all C*7*7 = 9408
// elements of that sample, so the whole sample must live in one workgroup).
//  pool(8x8 mean) -> LN0 -> depthwise 7x7 'SAME' conv on the 7x7 map
//  -> LN1 -> ReLU  => ker[b*C + c][49] written to workspace.
// conv outputs are recomputed in pass 2 instead of double-buffering LDS.
// ---------------------------------------------------------------------------
__global__ __launch_bounds__(256) void gen_kernels(
    const float* __restrict__ x,
    const float* __restrict__ ln0w, const float* __restrict__ ln0b,
    const float* __restrict__ cw,
    const float* __restrict__ ln1w, const float* __restrict__ ln1b,
    float* __restrict__ ker)
{
  __shared__ float q[PC];          // pooled / normalized map, 37.6 KB
  __shared__ float redA[256];
  __shared__ float redB[256];

  const int b = blockIdx.x;
  const int t = threadIdx.x;
  const float* xb = x + (size_t)b * CC * HH * WW;

  // ---- adaptive avg pool (8x8 uniform mean) ----
  float s = 0.f, s2 = 0.f;
  for (int v = t; v < PC; v += 256) {
    const int c = v / 49, r = v % 49, ky = r / 7, kx = r % 7;
    const float* xp = xb + ((size_t)c * HH + ky * 8) * WW + kx * 8;
    float acc = 0.f;
    #pragma unroll
    for (int i = 0; i < 8; ++i)
      #pragma unroll
      for (int j = 0; j < 8; ++j)
        acc += xp[i * WW + j];
    acc *= (1.f / 64.f);
    q[v] = acc;
    s += acc; s2 += acc * acc;
  }

  // ---- block reduction for LN0 stats ----
  redA[t] = s; redB[t] = s2;
  __syncthreads();
  for (int off = 128; off > 0; off >>= 1) {
    if (t < off) { redA[t] += redA[t + off]; redB[t] += redB[t + off]; }
    __syncthreads();
  }
  const float m0   = redA[0] * (1.f / PC);
  const float inv0 = rsqrtf(redB[0] * (1.f / PC) - m0 * m0 + LN_EPS);
  __syncthreads();   // WAR: everyone has read redA[0]/redB[0]

  // ---- LN0 (affine), in place ----
  for (int v = t; v < PC; v += 256)
    q[v] = (q[v] - m0) * inv0 * ln0w[v] + ln0b[v];
  __syncthreads();

  // ---- conv0 pass 1: stats of depthwise 7x7 'SAME' on the 7x7 map ----
  s = 0.f; s2 = 0.f;
  for (int v = t; v < PC; v += 256) {
    const int c = v / 49, r = v % 49, ky = r / 7, kx = r % 7;
    const float* wc = cw + c * 49;
    const float* qc = q  + c * 49;
    float acc = 0.f;
    for (int di = 0; di < 7; ++di) {
      const int yy = ky + di - 3;
      if (yy < 0 || yy >= 7) continue;
      for (int dj = 0; dj < 7; ++dj) {
        const int xx = kx + dj - 3;
        if (xx < 0 || xx >= 7) continue;
        acc += wc[di * 7 + dj] * qc[yy * 7 + xx];
      }
    }
    s += acc; s2 += acc * acc;
  }
  redA[t] = s; redB[t] = s2;
  __syncthreads();
  for (int off = 128; off > 0; off >>= 1) {
    if (t < off) { redA[t] += redA[t + off]; redB[t] += redB[t + off]; }
    __syncthreads();
  }
  const float m1   = redA[0] * (1.f / PC);
  const float inv1 = rsqrtf(redB[0] * (1.f / PC) - m1 * m1 + LN_EPS);
  __syncthreads();

  // ---- conv0 pass 2: recompute, LN1 + ReLU, emit dynamic kernels ----
  for (int v = t; v < PC; v += 256) {
    const int c = v / 49, r = v % 49, ky = r / 7, kx = r % 7;
    const float* wc = cw + c * 49;
    const float* qc = q  + c * 49;
    float acc = 0.f;
    for (int di = 0; di < 7; ++di) {
      const int yy = ky + di - 3;
      if (yy < 0 || yy >= 7) continue;
      for (int dj = 0; dj < 7; ++dj) {
        const int xx = kx + dj - 3;
        if (xx < 0 || xx >= 7) continue;
        acc += wc[di * 7 + dj] * qc[yy * 7 + xx];
      }
    }
    const float o = (acc - m1) * inv1 * ln1w[v] + ln1b[v];
    ker[(size_t)b * PC + v] = fmaxf(o, 0.f);
  }
}

// ---------------------------------------------------------------------------
// Kernel 2: dynamic depthwise 7x7 conv over the full image, via fp32 WMMA.
// One block (128 threads = 4 waves) per (b,c) channel.
//   out[y,x] = sum_dy sum_k A_dy[y,k] * B_dy[k,x]
//   A_dy[y,k] = img[y+dy-3, x0-3+k]   (shifted input rows, LDS)
//   B_dy[k,x] = w[dy, k-x] for 0<=k-x<=6 else 0  (banded Toeplitz, LDS)
// Each wave owns one 16-row output strip with FOUR 16x16 accumulators so a
// B-Toeplitz fragment is loaded once per (dy,ks) and reused by 4 back-to-back
// independent v_wmma_f32_16x16x4_f32 issues.
// ---------------------------------------------------------------------------
__global__ __launch_bounds__(128) void dyn_dwconv(
    const float* __restrict__ x,
    const float* __restrict__ ker,
    float* __restrict__ out)
{
  // Image + halo: LDS row r <-> image row r-3 (r in [0,70)), col c <-> image
  // col c-3; columns reach tx+23 = 71 for the last tile's K padding.
  __shared__ float img[70][IMG_W];            // 21.3 KB
  __shared__ float Bt[7][24][16];             // 10.5 KB Toeplitz per dy

  const int bc = blockIdx.x;                  // b*C + c
  const int t  = threadIdx.x;
  const float* xc = x   + (size_t)bc * HH * WW;
  const float* kc = ker + (size_t)bc * 49;

  for (int i = t; i < 70 * IMG_W; i += 128) {
    const int r = i / IMG_W, c = i % IMG_W;
    const int gy = r - 3, gx = c - 3;
    img[r][c] = (gy >= 0 && gy < HH && gx >= 0 && gx < WW)
                    ? xc[gy * WW + gx] : 0.f;
  }
  for (int i = t; i < 7 * 24 * 16; i += 128) {
    const int dy = i / (24 * 16), rem = i % (24 * 16);
    const int k = rem / 16, n = rem % 16;
    const int dx = k - n;
    Bt[dy][k][n] = (dx >= 0 && dx < 7) ? kc[dy * 7 + dx] : 0.f;
  }
  __syncthreads();

  const int lane = t & 31;
  const int wave = t >> 5;
  const int m  = lane & 15;        // A: row M / B,C,D: column N
  const int hi = lane >> 4;        // lane half
  const int ty = wave * 16;        // this wave's output row strip

  v8f acc[4] = {};                 // 4 column tiles (tx = 0,16,32,48)

  for (int dy = 0; dy < 7; ++dy) {
    const float* rowp = img[ty + m + dy];   // A row for this lane, this dy
    #pragma unroll
    for (int ks = 0; ks < 6; ++ks) {
      // 16x4 f32 A/B lane layout: lanes 0-15 hold K={0,1}, lanes 16-31 hold
      // K={2,3}; VGPR0 = even K of the pair, VGPR1 = odd K.
      const int kb = 4 * ks + 2 * hi;
      v2f bf = { Bt[dy][kb][m], Bt[dy][kb + 1][m] };   // load once per (dy,ks)
      #pragma unroll
      for (int tt = 0; tt < 4; ++tt) {                 // reuse across 4 tiles
        v2f a = { rowp[tt * 16 + kb], rowp[tt * 16 + kb + 1] };
        acc[tt] = __builtin_amdgcn_wmma_f32_16x16x4_f32(
            /*neg_a=*/false, a, /*neg_b=*/false, bf,
            /*c_mod=*/(short)0, acc[tt], /*reuse_a=*/false, /*reuse_b=*/false);
      }
    }
  }

  // C/D 16x16 f32 layout: VGPR r -> row ty+r (lanes 0-15) / ty+r+8 (16-31),
  // col tx + (lane&15).
  #pragma unroll
  for (int tt = 0; tt < 4; ++tt) {
    const int col = tt * 16 + m;
    if (col < WW) {
      #pragma unroll
      for (int r = 0; r < 8; ++r) {
        const int row = ty + r + hi * 8;
        if (row < HH)
          out[(size_t)bc * HH * WW + row * WW + col] = acc[tt][r];
      }
    }
  }
}

extern "C" void kernel_launch(void* const* d_in, const int* in_sizes, int n_in,
                              void* d_out, int out_size, void* d_ws, size_t ws_size,
                              hipStream_t stream) {
  (void)in_sizes; (void)n_in; (void)out_size; (void)ws_size;
  const float* x    = (const float*)d_in[0];
  const float* ln0w = (const float*)d_in[1];
  const float* ln0b = (const float*)d_in[2];
  const float* cw   = (const float*)d_in[3];
  const float* ln1w = (const float*)d_in[4];
  const float* ln1b = (const float*)d_in[5];
  float* ker = (float*)d_ws;                 // BB*PC floats = 1.2 MB scratch
  float* out = (float*)d_out;

  gen_kernels<<<BB, 256, 0, stream>>>(x, ln0w, ln0b, cw, ln1w, ln1b, ker);
  dyn_dwconv<<<BB * CC, 128, 0, stream>>>(x, ker, out);
}